// LOUPEPolicy_5669356833984
// MI455X (gfx1250) — compile-verified
//
#include <hip/hip_runtime.h>
#include <stdint.h>

#define W 368
#define BATCH 4
#define HALF_W 184
#define SLOPEF 10.0f
#define BUDGETF 92.0f
#define MAX_TRIES 2000

// masked_kspace geometry: B=4, M=16, H=640, W=368, C=2
#define N_KS 30146560
#define N4_KS 7536640       // float4 count
#define N4_PER_B 1884160    // float4s per batch
#define N4_PER_ROW 184      // float4s per (b,m,h) row (W*C/4)

typedef float v2f __attribute__((ext_vector_type(2)));
typedef float v8f __attribute__((ext_vector_type(8)));

// ---------------------------------------------------------------------------
// JAX threefry2x32 (rotations and key-schedule match jax._src.prng)
// ---------------------------------------------------------------------------
__device__ __forceinline__ void tf2x32(uint32_t k0, uint32_t k1,
                                       uint32_t c0, uint32_t c1,
                                       uint32_t& o0, uint32_t& o1) {
  const uint32_t ks[3] = {k0, k1, k0 ^ k1 ^ 0x1BD11BDAu};
  uint32_t x0 = c0 + ks[0], x1 = c1 + ks[1];
  const int R0[4] = {13, 15, 26, 6};
  const int R1[4] = {17, 29, 16, 24};
#pragma unroll
  for (int i = 0; i < 5; ++i) {
    const int* r = (i & 1) ? R1 : R0;
#pragma unroll
    for (int j = 0; j < 4; ++j) {
      x0 += x1;
      x1 = (x1 << r[j]) | (x1 >> (32 - r[j]));
      x1 ^= x0;
    }
    x0 += ks[(i + 1) % 3];
    x1 += ks[(i + 2) % 3] + (uint32_t)(i + 1);
  }
  o0 = x0;
  o1 = x1;
}

__device__ __forceinline__ float u01(uint32_t bits) {
  float f = __uint_as_float((bits >> 9) | 0x3F800000u) - 1.0f;
  return fmaxf(f, 0.0f);
}

// ---------------------------------------------------------------------------
// Kernel 1: full policy on (B=4, W=368). One block, 4 waves; wave b owns row b.
// Uses v_wmma_f32_16x16x4_f32 as the cross-wave reduction for count / xbar.
// ---------------------------------------------------------------------------
__global__ __launch_bounds__(128) void loupe_policy(
    const float* __restrict__ mask, const float* __restrict__ sampler,
    float* __restrict__ out_newmask, float* __restrict__ out_prob) {
  __shared__ float s_mp[BATCH][W];
  __shared__ float s_part[2][BATCH][4];
  __shared__ float s_res8[8];

  const int tid = threadIdx.x;
  const int b = tid >> 5;
  const int lane = tid & 31;

  // ---- Phase A: softplus prob + row max of (1-mask)*pm ----
  float pmv[12];
  float mx = -1e30f;
#pragma unroll
  for (int j = 0; j < 12; ++j) {
    int w = lane + 32 * j;
    float pm = 0.0f, val = -1e30f;
    if (w < W) {
      float x = SLOPEF * sampler[w];
      float sp = fmaxf(x, 0.0f) + log1pf(expf(-fabsf(x)));
      pm = sp / SLOPEF;
      float mv = mask[b * W + w];
      val = (1.0f - mv) * pm;
    }
    pmv[j] = pm;
    mx = fmaxf(mx, val);
  }
  for (int off = 16; off >= 1; off >>= 1)
    mx = fmaxf(mx, __shfl_xor(mx, off, 32));
  const float denom = mx;

  // ---- Phase B: masked prob, accumulate count & sum(masked_prob) ----
  float cnt = 0.0f, xs = 0.0f;
#pragma unroll
  for (int j = 0; j < 12; ++j) {
    int w = lane + 32 * j;
    if (w < W) {
      float mv = mask[b * W + w];
      float m0 = (mv == 0.0f) ? 1.0f : 0.0f;
      float mp = (pmv[j] / denom) * (1.0f - mv);
      s_mp[b][w] = mp;
      cnt += m0;
      xs += mp * m0;
    }
  }
  // fold 32 lanes -> 4 group partials (lanes 0..3 hold sums of lanes l%4==g)
  for (int off = 16; off >= 4; off >>= 1) {
    cnt += __shfl_xor(cnt, off, 32);
    xs += __shfl_xor(xs, off, 32);
  }
  if (lane < 4) {
    s_part[0][b][lane] = cnt;  // rows 0..3 of A: count partials per batch
    s_part[1][b][lane] = xs;   // rows 4..7 of A: xbar-sum partials per batch
  }
  __syncthreads();

  // ---- WMMA reduction: D = A(16x4 partials) x B(4x16 ones) ----
  if (tid < 32) {
    int rm = lane & 15;
    v2f a;
    a.x = 0.0f;
    a.y = 0.0f;
    if (rm < 8) {
      int q = rm >> 2, bb = rm & 3;
      int k0 = (lane >= 16) ? 2 : 0;  // VGPR0: K=0/2, VGPR1: K=1/3
      a.x = s_part[q][bb][k0];
      a.y = s_part[q][bb][k0 + 1];
    }
    v2f bones;
    bones.x = 1.0f;
    bones.y = 1.0f;
    v8f c = {};
    v8f d = __builtin_amdgcn_wmma_f32_16x16x4_f32(false, a, false, bones,
                                                  (short)0, c, false, false);
    if (lane == 0) {  // lane 0 (N=0), VGPR m holds D[m][0] = row-m sum
#pragma unroll
      for (int m = 0; m < 8; ++m) s_res8[m] = d[m];
    }
  }
  __syncthreads();

  const float count = s_res8[b];
  const float xsum = s_res8[4 + b];
  const float sparsity = BUDGETF / count;
  const float xbar0 = xsum / count;
  const float r = sparsity / xbar0;
  const float beta = (1.0f - sparsity) / (1.0f - xbar0);
  const bool le = (r <= 1.0f);

  // ---- rescale_probs + emit final_prob_mask + row mean for rejection test --
  float rowsum = 0.0f;
#pragma unroll
  for (int j = 0; j < 12; ++j) {
    int w = lane + 32 * j;
    if (w < W) {
      float mv = mask[b * W + w];
      float mp = s_mp[b][w];
      float resc = le ? mp * r : 1.0f - (1.0f - mp) * beta;
      float mp2 = (mv == 0.0f) ? resc : mp;
      s_mp[b][w] = mp2;
      out_prob[b * W + w] = mp2;
      rowsum += mp2;
    }
  }
  for (int off = 16; off >= 1; off >>= 1)
    rowsum += __shfl_xor(rowsum, off, 32);
  const float xbar_row = rowsum / (float)W;
  __syncthreads();

  // ---- binarize: replicate jax.random.split(key(42), 4) -> keys[b] ----
  // split(root,4): counts 0..7, pairs (i,i+4) -> (a_i,b_i);
  // keys = [(a0,a1),(a2,a3),(b0,b1),(b2,b3)]
  uint32_t A0, B0, A1, B1;
  uint32_t cb = (uint32_t)((b & 1) * 2);
  tf2x32(0u, 42u, cb, cb + 4u, A0, B0);
  tf2x32(0u, 42u, cb + 1u, cb + 5u, A1, B1);
  uint32_t key0 = (b < 2) ? A0 : B0;
  uint32_t key1 = (b < 2) ? A1 : B1;

  float res1[6], res2[6];
#pragma unroll
  for (int j = 0; j < 6; ++j) {
    res1[j] = 0.0f;
    res2[j] = 0.0f;
  }

  int tries = 0;
  while (true) {
    // key, sub = split(key): counts [0,1,2,3], pairs (0,2),(1,3)
    uint32_t s0, s1, t0, t1;
    tf2x32(key0, key1, 0u, 2u, s0, t0);
    tf2x32(key0, key1, 1u, 3u, s1, t1);
    key0 = s0;
    key1 = s1;
    const uint32_t sub0 = t0, sub1 = t1;

    float rs = 0.0f;
#pragma unroll
    for (int j = 0; j < 6; ++j) {
      int i = lane + 32 * j;
      float r1 = 0.0f, r2 = 0.0f;
      if (i < HALF_W) {
        uint32_t oA, oB;
        tf2x32(sub0, sub1, (uint32_t)i, (uint32_t)(i + HALF_W), oA, oB);
        r1 = (s_mp[b][i] > u01(oA)) ? 1.0f : 0.0f;
        r2 = (s_mp[b][i + HALF_W] > u01(oB)) ? 1.0f : 0.0f;
        rs += r1 + r2;
      }
      res1[j] = r1;
      res2[j] = r2;
    }
    for (int off = 16; off >= 1; off >>= 1)
      rs += __shfl_xor(rs, off, 32);
    float mean = rs / (float)W;
    ++tries;
    bool ok = fabsf(mean - xbar_row) <= (1e-3f + 1e-5f * fabsf(xbar_row));
    if (ok || tries >= MAX_TRIES) break;
  }

  // new_mask = mask + accepted binary draw
#pragma unroll
  for (int j = 0; j < 6; ++j) {
    int i = lane + 32 * j;
    if (i < HALF_W) {
      out_newmask[b * W + i] = mask[b * W + i] + res1[j];
      out_newmask[b * W + i + HALF_W] = mask[b * W + i + HALF_W] + res2[j];
    }
  }
}

// ---------------------------------------------------------------------------
// Kernel 2: masked_kspace = new_mask * kspace (+ sign fix), HBM-bound stream.
// Stages the 5.9KB new_mask table into LDS with the CDNA5 async data mover
// (global_load_async_to_lds_b128, ASYNCcnt), then one float4 per thread.
// ---------------------------------------------------------------------------
__global__ __launch_bounds__(256) void loupe_apply(
    const float* __restrict__ kspace, const float* __restrict__ newmask,
    float* __restrict__ out) {
  __shared__ float s_mask[BATCH * W];  // 1472 floats = 368 x b128 chunks

  for (int c = threadIdx.x; c < (BATCH * W) / 4; c += (int)blockDim.x) {
    uint32_t laddr = (uint32_t)(uintptr_t)(&s_mask[4 * c]);
    uint64_t gaddr = (uint64_t)(uintptr_t)(newmask + 4 * c);
    asm volatile("global_load_async_to_lds_b128 %0, %1, off" ::"v"(laddr),
                 "v"(gaddr)
                 : "memory");
  }
  asm volatile("s_wait_asynccnt 0x0" ::: "memory");
  __syncthreads();

  int idx = (int)(blockIdx.x * blockDim.x + threadIdx.x);
  if (idx < N4_KS) {
    int b = idx / N4_PER_B;
    int rem = idx % N4_PER_ROW;  // float4 within a (b,m,h) row
    int w0 = rem * 2;            // covers columns w0 (x,y) and w0+1 (z,w)
    const float4 k = ((const float4*)kspace)[idx];
    float m0 = s_mask[b * W + w0];
    float m1 = s_mask[b * W + w0 + 1];
    float4 o;
    o.x = m0 * k.x;
    if (k.x < 0.0f && m0 == 0.0f) o.x = -o.x;
    o.y = m0 * k.y;
    if (k.y < 0.0f && m0 == 0.0f) o.y = -o.y;
    o.z = m1 * k.z;
    if (k.z < 0.0f && m1 == 0.0f) o.z = -o.z;
    o.w = m1 * k.w;
    if (k.w < 0.0f && m1 == 0.0f) o.w = -o.w;
    ((float4*)out)[idx] = o;
  }
}

// ---------------------------------------------------------------------------
extern "C" void kernel_launch(void* const* d_in, const int* in_sizes, int n_in,
                              void* d_out, int out_size, void* d_ws,
                              size_t ws_size, hipStream_t stream) {
  (void)in_sizes;
  (void)n_in;
  (void)out_size;
  (void)d_ws;
  (void)ws_size;
  const float* mask = (const float*)d_in[0];
  const float* kspace = (const float*)d_in[1];
  const float* sampler = (const float*)d_in[2];

  float* out = (float*)d_out;
  float* out_newmask = out;                  // [0, 1472)
  float* out_kspace = out + BATCH * W;       // [1472, 1472+30146560)
  float* out_prob = out + BATCH * W + N_KS;  // last 1472

  loupe_policy<<<1, 128, 0, stream>>>(mask, sampler, out_newmask, out_prob);

  int blocks = (N4_KS + 255) / 256;  // 29440
  loupe_apply<<<blocks, 256, 0, stream>>>(kspace, out_newmask, out_kspace);
}